// CNN_88098369175791
// MI455X (gfx1250) — compile-verified
//
#include <hip/hip_runtime.h>
#include <math.h>

// ---------------------------------------------------------------------------
// Fused single-workgroup implementation of the reference network for gfx1250.
// All GEMMs run on V_WMMA_F32_16X16X4_F32 (fp32 in/out, matches reference
// precision). One block of 256 threads (8 wave32 waves); all intermediates in
// LDS (~150 KB of the 320 KB/WGP); barriers between phases.
// ---------------------------------------------------------------------------

typedef __attribute__((ext_vector_type(2))) float v2f;
typedef __attribute__((ext_vector_type(8))) float v8f;

#define NT 256   // threads per block
#define NW 8     // waves per block (wave32)
#define LD 128   // padded row stride for 119-wide matrices

static __device__ __forceinline__ int imin(int a, int b) { return a < b ? a : b; }
static __device__ __forceinline__ float sigmf(float x) { return 1.0f / (1.0f + expf(-x)); }

// -------- LDS arena offsets (floats). Zeroed once at kernel start so all
// padding regions are guaranteed zero; stores are masked to valid regions. ---
constexpr int OF_EEGQ = 0;                  // [16][LD] eeg_q
constexpr int OF_KVS  = OF_EEGQ + 16 * LD;  // [16][LD] sliding-window K/V source
constexpr int OF_QP   = OF_KVS  + 16 * LD;  // [16][LD] qp  -> later wA_p
constexpr int OF_KP   = OF_QP   + 16 * LD;  // [16][LD] kp  -> later wB_p
constexpr int OF_VP   = OF_KP   + 16 * LD;  // [16][LD] vp  -> later eeg_ln
constexpr int OF_AV   = OF_VP   + 16 * LD;  // [16][LD] attnV -> later qc / av_c ([128][16])
constexpr int OF_ATTA = OF_AV   + 16 * LD;  // [16][LD] wA_att -> later kc ([128][16])
constexpr int OF_ATTB = OF_ATTA + 16 * LD;  // [16][LD] wB_att -> later vc ([128][16])
constexpr int OF_S    = OF_ATTB + 16 * LD;  // [16][16] TD attention scores
constexpr int OF_G    = OF_S    + 256;      // [16][16] eeg_q @ att^T
constexpr int OF_WAVA = OF_G    + 256;      // [160] wavA (padded)
constexpr int OF_WAVB = OF_WAVA + 160;      // [160] wavB (padded)
constexpr int OF_ATTN = OF_WAVB + 160;      // [128][128] CM attention matrix
constexpr int OF_R0   = OF_ATTN + 128*128;  // [128][16] outs (even i)
constexpr int OF_R1   = OF_R0   + 128*16;   // [128][16] outs (odd i)
constexpr int OF_VAB  = OF_R1   + 128*16;   // [32] vA ++ vB
constexpr int OF_WVEC = OF_VAB  + 32;       // [16] weight vector
constexpr int OF_DD   = OF_WVEC + 16;       // [256] dd = s0 ++ s1
constexpr int OF_H    = OF_DD   + 256;      // [128] hidden
constexpr int SMEMF   = OF_H    + 128;      // total floats (~149 KB)

// -------- WMMA helpers: D(16x16) += A(16x4) * B(4x16) over K4 chunks -------
// A layout: lane l<16 holds row l, K = {4k, 4k+1}; lanes 16-31 hold K = {4k+2, 4k+3}
// B layout: lane l<16 holds col l, same K split. C/D: VGPR v = rows {v, v+8}.
template <typename LA, typename LB>
static __device__ __forceinline__ v8f wmma_tile(int K4, LA&& la, LB&& lb) {
  const int lane = threadIdx.x & 31;
  const int l16  = lane & 15;
  const int kh   = (lane >> 4) << 1;   // 0 or 2
  v8f acc = {0.f, 0.f, 0.f, 0.f, 0.f, 0.f, 0.f, 0.f};
  for (int k4 = 0; k4 < K4; ++k4) {
    const int kb = (k4 << 2) + kh;
    v2f a, b;
    a.x = la(l16, kb);     a.y = la(l16, kb + 1);
    b.x = lb(kb, l16);     b.y = lb(kb + 1, l16);
    acc = __builtin_amdgcn_wmma_f32_16x16x4_f32(false, a, false, b,
                                                (short)0, acc, false, false);
  }
  return acc;
}

template <typename FS>
static __device__ __forceinline__ void wmma_store(const v8f& acc, FS&& st) {
  const int lane = threadIdx.x & 31;
  const int col  = lane & 15;
  const int rb   = (lane >> 4) << 3;
#pragma unroll
  for (int r = 0; r < 8; ++r) st(rb + r, col, acc[r]);
}

__global__ __launch_bounds__(NT) void fused_net_gfx1250(
    const float* __restrict__ x,
    const float* __restrict__ td_in_w,  const float* __restrict__ td_in_b,
    const float* __restrict__ td_out_w, const float* __restrict__ td_out_b,
    const float* __restrict__ cm_in_w,  const float* __restrict__ cm_in_b,
    const float* __restrict__ cm_out_w, const float* __restrict__ cm_out_b,
    const float* __restrict__ mc_w,     const float* __restrict__ mc_b,
    const float* __restrict__ max_fc_w, const float* __restrict__ max_fc_b,
    const float* __restrict__ proj_w,
    const float* __restrict__ ln_g,     const float* __restrict__ ln_b,
    const float* __restrict__ fc_w,     const float* __restrict__ fc_b,
    const float* __restrict__ out1_w,   const float* __restrict__ out1_b,
    const float* __restrict__ out2_w,   const float* __restrict__ out2_b,
    float* __restrict__ out) {
  __shared__ float sm[SMEMF];
  __shared__ int s_mi;
  const int tid = threadIdx.x;
  const int wid = tid >> 5;

  // ---- Phase 0: zero the LDS arena (establishes all zero-padding) ----------
  for (int i = tid; i < SMEMF; i += NT) sm[i] = 0.f;
  __syncthreads();

  // ---- Phase 1: stage inputs ----------------------------------------------
  for (int t = tid; t < 140; t += NT) {
    sm[OF_WAVA + t] = x[t];
    sm[OF_WAVB + t] = x[17 * 140 + t];
  }
  for (int i = tid; i < 16 * 119; i += NT) {
    const int c = i / 119, t = i % 119;
    sm[OF_EEGQ + c * LD + t] = x[(1 + c) * 140 + 21 + t];   // eeg[:, 21:140]
  }
  __syncthreads();

  // ---- Phase 2: qp = eeg_q @ Wq^T + bq  (16 x 119), shared by both wavs ----
  for (int nt = wid; nt < 8; nt += NW) {
    const int n0 = nt * 16;
    v8f acc = wmma_tile(30,
      [&](int m, int k) { return sm[OF_EEGQ + m * LD + k]; },
      [&](int k, int n) {
        const int nn = imin(n0 + n, 118), kk = imin(k, 118);
        return td_in_w[nn * 119 + kk];
      });
    wmma_store(acc, [&](int m, int c, float v) {
      const int n = n0 + c;
      if (n < 119) sm[OF_QP + m * LD + n] = v + td_in_b[n];
    });
  }
  __syncthreads();

  // ---- Phase 3: time-delay MHA for wavA then wavB --------------------------
  const float td_scale = 1.0f / sqrtf(119.0f);
  for (int wv = 0; wv < 2; ++wv) {
    const float* wav = &sm[wv ? OF_WAVB : OF_WAVA];
    float* att = &sm[wv ? OF_ATTB : OF_ATTA];

    // K/V source: (14 x 119), src[m][k] = wav[k+m]
    for (int i = tid; i < 14 * 119; i += NT) {
      const int m = i / 119, k = i % 119;
      sm[OF_KVS + m * LD + k] = wav[k + m];
    }
    __syncthreads();

    // kp, vp: (14 x 119) projections (16 tiles: p in {k,v} x 8 n-tiles)
    for (int tt = wid; tt < 16; tt += NW) {
      const int p = tt >> 3;            // 0 = k-proj, 1 = v-proj
      const int n0 = (tt & 7) * 16;
      const int wo = p ? 238 : 119;     // weight/bias row offset
      v8f acc = wmma_tile(30,
        [&](int m, int k) { return sm[OF_KVS + m * LD + k]; },
        [&](int k, int n) {
          const int nn = imin(n0 + n, 118), kk = imin(k, 118);
          return td_in_w[(wo + nn) * 119 + kk];
        });
      float* dst = &sm[p ? OF_VP : OF_KP];
      wmma_store(acc, [&](int m, int c, float v) {
        const int n = n0 + c;
        if (m < 14 && n < 119) dst[m * LD + n] = v + td_in_b[wo + n];
      });
    }
    __syncthreads();

    // scores S = qp @ kp^T (16 x 14), single tile on wave 0
    if (wid == 0) {
      v8f acc = wmma_tile(30,
        [&](int m, int k) { return sm[OF_QP + m * LD + k]; },
        [&](int k, int n) { return sm[OF_KP + n * LD + k]; });
      wmma_store(acc, [&](int m, int c, float v) {
        if (c < 14) sm[OF_S + m * 16 + c] = v;
      });
    }
    __syncthreads();

    // softmax over 14 keys per row (scale applied pre-softmax)
    if (tid < 16) {
      float* row = &sm[OF_S + tid * 16];
      float mx = -1e30f;
      for (int j = 0; j < 14; ++j) mx = fmaxf(mx, row[j] * td_scale);
      float s = 0.f;
      for (int j = 0; j < 14; ++j) { const float e = expf(row[j] * td_scale - mx); row[j] = e; s += e; }
      const float inv = 1.f / s;
      for (int j = 0; j < 14; ++j) row[j] *= inv;
    }
    __syncthreads();

    // attnV = S @ vp  (16 x 119); S cols 14,15 are zero (padding)
    for (int nt = wid; nt < 8; nt += NW) {
      const int n0 = nt * 16;
      v8f acc = wmma_tile(4,
        [&](int m, int k) { return sm[OF_S + m * 16 + k]; },
        [&](int k, int n) { return sm[OF_VP + k * LD + n0 + n]; });
      wmma_store(acc, [&](int m, int c, float v) {
        const int n = n0 + c;
        if (n < 119) sm[OF_AV + m * LD + n] = v;
      });
    }
    __syncthreads();

    // att = attnV @ out_w^T + out_b  (16 x 119)
    for (int nt = wid; nt < 8; nt += NW) {
      const int n0 = nt * 16;
      v8f acc = wmma_tile(30,
        [&](int m, int k) { return sm[OF_AV + m * LD + k]; },
        [&](int k, int n) {
          const int nn = imin(n0 + n, 118), kk = imin(k, 118);
          return td_out_w[nn * 119 + kk];
        });
      wmma_store(acc, [&](int m, int c, float v) {
        const int n = n0 + c;
        if (n < 119) att[m * LD + n] = v + td_out_b[n];
      });
    }
    __syncthreads();

    // G = eeg_q @ att^T (16 x 16), wave 0
    if (wid == 0) {
      v8f acc = wmma_tile(30,
        [&](int m, int k) { return sm[OF_EEGQ + m * LD + k]; },
        [&](int k, int n) { return att[n * LD + k]; });
      wmma_store(acc, [&](int m, int c, float v) { sm[OF_G + m * 16 + c] = v; });
    }
    __syncthreads();

    // v{A,B}[n] = relu(mc_w[wv] . G[:, n] + mc_b[wv])
    if (tid < 16) {
      float s = mc_b[wv];
      for (int m = 0; m < 16; ++m) s += mc_w[wv * 16 + m] * sm[OF_G + m * 16 + tid];
      sm[OF_VAB + wv * 16 + tid] = fmaxf(s, 0.f);
    }
    __syncthreads();
  }

  // ---- Phase 4: gating weight + argmax ------------------------------------
  if (tid < 16) {
    float s = max_fc_b[tid];
    for (int i = 0; i < 32; ++i) s += max_fc_w[tid * 32 + i] * sm[OF_VAB + i];
    sm[OF_WVEC + tid] = fmaxf(s, 0.f);
  }
  __syncthreads();
  if (tid == 0) {
    int mi = 0; float best = sm[OF_WVEC];
    for (int j = 1; j < 16; ++j) {
      const float v = sm[OF_WVEC + j];
      if (v > best) { best = v; mi = j; }
    }
    s_mi = imin(mi, 13);   // jnp.take default 'clip' (wA_win.T has 14 rows)
  }
  __syncthreads();

  // ---- Phase 5: wA_p / wB_p projections + layernorm ------------------------
  {
    const int mi = s_mi;
    for (int i = tid; i < 16 * 119; i += NT) {
      const int c = i / 119, t = i % 119;
      sm[OF_QP + c * LD + t] = proj_w[c]      * sm[OF_WAVA + t + mi];  // wA_p
      sm[OF_KP + c * LD + t] = proj_w[16 + c] * sm[OF_WAVB + t + mi];  // wB_p
    }
  }
  for (int t = tid; t < 119; t += NT) {
    float mu = 0.f;
    for (int c = 0; c < 16; ++c) mu += sm[OF_EEGQ + c * LD + t];
    mu *= (1.f / 16.f);
    float var = 0.f;
    for (int c = 0; c < 16; ++c) { const float d = sm[OF_EEGQ + c * LD + t] - mu; var += d * d; }
    var *= (1.f / 16.f);
    const float inv = rsqrtf(var + 1e-5f);
    for (int c = 0; c < 16; ++c)
      sm[OF_VP + c * LD + t] = (sm[OF_EEGQ + c * LD + t] - mu) * inv * ln_g[c] + ln_b[c]; // eeg_ln
  }
  __syncthreads();

  // ---- Phase 6: four cross-modal MHAs (E=16, 119 tokens) -------------------
  // matrices live in: 0 -> wA_p(OF_QP), 1 -> wB_p(OF_KP), 2 -> eeg_ln(OF_VP)
  const int dOff[4] = {OF_QP, OF_VP, OF_VP, OF_KP};   // data[i]
  const int kOff[4] = {OF_VP, OF_QP, OF_KP, OF_VP};   // kv[i]
  for (int ci = 0; ci < 4; ++ci) {
    const float* dmat = &sm[dOff[ci]];
    const float* kmat = &sm[kOff[ci]];
    const float* inw = cm_in_w + ci * 48 * 16;
    const float* inb = cm_in_b + ci * 48;

    // qc/kc/vc: (119 x 16) each; 24 tiles (3 proj x 8 m-tiles)
    for (int tt = wid; tt < 24; tt += NW) {
      const int p = tt / 8;               // 0=q (from data), 1=k, 2=v (from kv)
      const int m0 = (tt % 8) * 16;
      const float* src = (p == 0) ? dmat : kmat;
      float* dst = &sm[(p == 0) ? OF_AV : (p == 1) ? OF_ATTA : OF_ATTB];
      v8f acc = wmma_tile(4,
        [&](int m, int k) { return src[k * LD + m0 + m]; },   // A = src^T
        [&](int k, int n) { return inw[(p * 16 + n) * 16 + k]; });
      wmma_store(acc, [&](int m, int c, float v) {
        const int mm = m0 + m;
        if (mm < 119) dst[mm * 16 + c] = v + inb[p * 16 + c];
      });
    }
    __syncthreads();

    // attn = qc @ kc^T (119 x 119): 64 tiles
    for (int tt = wid; tt < 64; tt += NW) {
      const int m0 = (tt >> 3) * 16, n0 = (tt & 7) * 16;
      v8f acc = wmma_tile(4,
        [&](int m, int k) { return sm[OF_AV + (m0 + m) * 16 + k]; },
        [&](int k, int n) { return sm[OF_ATTA + (n0 + n) * 16 + k]; });
      wmma_store(acc, [&](int m, int c, float v) {
        const int mm = m0 + m, nn = n0 + c;
        if (mm < 119 && nn < 119) sm[OF_ATTN + mm * LD + nn] = v;
      });
    }
    __syncthreads();

    // softmax rows (scale = 16^-0.5 = 0.25)
    for (int r = tid; r < 119; r += NT) {
      float* row = &sm[OF_ATTN + r * LD];
      float mx = -1e30f;
      for (int j = 0; j < 119; ++j) mx = fmaxf(mx, row[j] * 0.25f);
      float s = 0.f;
      for (int j = 0; j < 119; ++j) { const float e = expf(row[j] * 0.25f - mx); row[j] = e; s += e; }
      const float inv = 1.f / s;
      for (int j = 0; j < 119; ++j) row[j] *= inv;
    }
    __syncthreads();

    // av = attn @ vc (119 x 16) -> reuse qc slot
    for (int mt = wid; mt < 8; mt += NW) {
      const int m0 = mt * 16;
      v8f acc = wmma_tile(30,
        [&](int m, int k) { return sm[OF_ATTN + (m0 + m) * LD + k]; },
        [&](int k, int n) { return sm[OF_ATTB + k * 16 + n]; });
      wmma_store(acc, [&](int m, int c, float v) {
        const int mm = m0 + m;
        if (mm < 119) sm[OF_AV + mm * 16 + c] = v;
      });
    }
    __syncthreads();

    // res = av @ out_w^T + out_b (119 x 16)
    float* res = &sm[(ci & 1) ? OF_R1 : OF_R0];
    const float* ow = cm_out_w + ci * 16 * 16;
    const float* ob = cm_out_b + ci * 16;
    for (int mt = wid; mt < 8; mt += NW) {
      const int m0 = mt * 16;
      v8f acc = wmma_tile(4,
        [&](int m, int k) { return sm[OF_AV + (m0 + m) * 16 + k]; },
        [&](int k, int n) { return ow[n * 16 + k]; });
      wmma_store(acc, [&](int m, int c, float v) {
        const int mm = m0 + m;
        if (mm < 119) res[mm * 16 + c] = v + ob[c];
      });
    }
    __syncthreads();

    // per-token dots + sigmoid after each pair
    if (ci == 1 || ci == 3) {
      const int half = (ci == 3) ? 119 : 0;
      const int fi = (ci == 3) ? 1 : 0;
      const float fw = fc_w[fi], fb = fc_b[fi];
      for (int t = tid; t < 119; t += NT) {
        float d = 0.f;
        for (int c = 0; c < 16; ++c) d += sm[OF_R0 + t * 16 + c] * sm[OF_R1 + t * 16 + c];
        sm[OF_DD + half + t] = sigmf(fw * d + fb);
      }
      __syncthreads();
    }
  }

  // ---- Phase 7: dense heads ------------------------------------------------
  for (int j = tid; j < 119; j += NT) {
    float s = out1_b[j];
    for (int i = 0; i < 238; ++i) s += out1_w[j * 238 + i] * sm[OF_DD + i];
    sm[OF_H + j] = sigmf(s);
  }
  __syncthreads();
  if (tid < 2) {
    float s = out2_b[tid];
    for (int i = 0; i < 119; ++i) s += out2_w[tid * 119 + i] * sm[OF_H + i];
    out[tid] = sigmf(s);
  }
}

extern "C" void kernel_launch(void* const* d_in, const int* in_sizes, int n_in,
                              void* d_out, int out_size, void* d_ws, size_t ws_size,
                              hipStream_t stream) {
  (void)in_sizes; (void)n_in; (void)out_size; (void)d_ws; (void)ws_size;
  fused_net_gfx1250<<<1, NT, 0, stream>>>(
      (const float*)d_in[0],  (const float*)d_in[1],  (const float*)d_in[2],
      (const float*)d_in[3],  (const float*)d_in[4],  (const float*)d_in[5],
      (const float*)d_in[6],  (const float*)d_in[7],  (const float*)d_in[8],
      (const float*)d_in[9],  (const float*)d_in[10], (const float*)d_in[11],
      (const float*)d_in[12], (const float*)d_in[13], (const float*)d_in[14],
      (const float*)d_in[15], (const float*)d_in[16], (const float*)d_in[17],
      (const float*)d_in[18], (const float*)d_in[19], (const float*)d_in[20],
      (const float*)d_in[21], (float*)d_out);
}